// FoRefLoss_40063454937361
// MI455X (gfx1250) — compile-verified
//
#include <hip/hip_runtime.h>
#include <math.h>

// ---------------------------------------------------------------------------
// FoRefLoss on MI455X (gfx1250, wave32).
// Memory-bound streaming reduction: 201 MB in / 4 B out  ->  ~8.6 us floor
// at 23.3 TB/s. Strategy: b128 nontemporal loads (4 rows / 12 floats per
// thread per iter), fp32 accumulation, deterministic 2-pass reduction with
// V_WMMA_F32_16X16X4_F32 used as a single-instruction 32-lane fold.
// ---------------------------------------------------------------------------

typedef __attribute__((ext_vector_type(4))) float f4;
typedef __attribute__((ext_vector_type(2))) float v2f;
typedef __attribute__((ext_vector_type(8))) float v8f;

#define PI_F      3.14159265358979323846f
#define TWOPI_F   6.28318530717958647692f
#define INV_PI_F  0.31830988618379067154f

// Full 32-lane sum, returned in every lane.
// A-matrix layout (16x4 f32, 2 VGPRs): VGPR0 holds K=0 (lanes 0-15) and
// K=2 (lanes 16-31); VGPR1 (K=1/K=3) is zeroed. B = ones(4x16).
// => D[m][n] = p[m] + p[m+16] for all n.
// D layout: VGPR j, lanes 0-15 -> M=j; lanes 16-31 -> M=j+8.
// Lane<16 rowsum over 8 VGPRs = sum(p[0..7]+p[16..23]); lane>=16 = the rest.
// One shfl_xor(16) completes the reduction.
__device__ __forceinline__ float wave_reduce_wmma(float x) {
    v2f a; a[0] = x;    a[1] = 0.0f;
    v2f b; b[0] = 1.0f; b[1] = 1.0f;
    v8f c = {};
    c = __builtin_amdgcn_wmma_f32_16x16x4_f32(
        /*neg_a=*/false, a, /*neg_b=*/false, b,
        /*c_mod=*/(short)0, c, /*reuse_a=*/false, /*reuse_b=*/false);
    float s = c[0] + c[1] + c[2] + c[3] + c[4] + c[5] + c[6] + c[7];
    s += __shfl_xor(s, 16, 32);
    return s;
}

// Angle-channel contribution, matching the reference op order:
// a = |p*2pi - l*2pi| mod 2pi; if a > pi: a = 2pi - a; t = a/pi; return t*t
__device__ __forceinline__ float ang_term(float p, float l) {
    float pa = p * TWOPI_F;
    float la = l * TWOPI_F;
    float ad = fabsf(pa - la);
    float m  = fmodf(ad, TWOPI_F);
    float cl = (m > PI_F) ? (TWOPI_F - m) : m;
    float t  = cl * INV_PI_F;
    return t * t;
}

__global__ void loss_partial_kernel(const float* __restrict__ pred,
                                    const float* __restrict__ lab,
                                    float* __restrict__ partials,
                                    int ngroups,   // rows / 4
                                    int tailRows,  // rows % 4
                                    long rows) {
    const f4* __restrict__ p4 = reinterpret_cast<const f4*>(pred);
    const f4* __restrict__ l4 = reinterpret_cast<const f4*>(lab);

    float acc = 0.0f;
    const int stride = gridDim.x * blockDim.x;
    for (int g = blockIdx.x * blockDim.x + threadIdx.x; g < ngroups; g += stride) {
        const int b = g * 3;  // f4 index; 3 x b128 per array = 4 rows
        f4 pa0 = __builtin_nontemporal_load(&p4[b + 0]);
        f4 pa1 = __builtin_nontemporal_load(&p4[b + 1]);
        f4 pa2 = __builtin_nontemporal_load(&p4[b + 2]);
        f4 la0 = __builtin_nontemporal_load(&l4[b + 0]);
        f4 la1 = __builtin_nontemporal_load(&l4[b + 1]);
        f4 la2 = __builtin_nontemporal_load(&l4[b + 2]);
        // group layout: [x y A x][y A x y][A x y A]  (A = angle channel)
        float d;
        d = pa0[0] - la0[0]; acc += d * d;
        d = pa0[1] - la0[1]; acc += d * d;
        acc += ang_term(pa0[2], la0[2]);
        d = pa0[3] - la0[3]; acc += d * d;
        d = pa1[0] - la1[0]; acc += d * d;
        acc += ang_term(pa1[1], la1[1]);
        d = pa1[2] - la1[2]; acc += d * d;
        d = pa1[3] - la1[3]; acc += d * d;
        acc += ang_term(pa2[0], la2[0]);
        d = pa2[1] - la2[1]; acc += d * d;
        d = pa2[2] - la2[2]; acc += d * d;
        acc += ang_term(pa2[3], la2[3]);
    }

    // Scalar tail (rows % 4), handled once by thread 0 of block 0.
    if (blockIdx.x == 0 && threadIdx.x == 0 && tailRows > 0) {
        for (long r = rows - tailRows; r < rows; ++r) {
            const long base = r * 3;
            float d0 = pred[base + 0] - lab[base + 0];
            float d1 = pred[base + 1] - lab[base + 1];
            acc += d0 * d0 + d1 * d1;
            acc += ang_term(pred[base + 2], lab[base + 2]);
        }
    }

    // Control flow has reconverged: EXEC is all 1s here (WMMA requirement).
    float wsum = wave_reduce_wmma(acc);

    __shared__ float sw[8];  // up to 8 waves per 256-thread block
    const int lane = threadIdx.x & 31;
    const int wid  = threadIdx.x >> 5;
    if (lane == 0) sw[wid] = wsum;
    __syncthreads();
    if (threadIdx.x == 0) {
        float bsum = 0.0f;
        const int nw = blockDim.x >> 5;
        for (int i = 0; i < nw; ++i) bsum += sw[i];
        partials[blockIdx.x] = bsum;
    }
}

// Single wave (32 threads): deterministic fixed-order fold of block partials,
// then one WMMA lane-fold. EXEC all ones by construction.
__global__ void loss_final_kernel(const float* __restrict__ partials, int n,
                                  float* __restrict__ out) {
    float acc = 0.0f;
    for (int i = threadIdx.x; i < n; i += 32) acc += partials[i];
    float tot = wave_reduce_wmma(acc);
    if (threadIdx.x == 0) out[0] = tot;
}

extern "C" void kernel_launch(void* const* d_in, const int* in_sizes, int n_in,
                              void* d_out, int out_size, void* d_ws, size_t ws_size,
                              hipStream_t stream) {
    const float* pred = (const float*)d_in[0];
    const float* lab  = (const float*)d_in[1];
    float* out        = (float*)d_out;
    float* partials   = (float*)d_ws;

    const long rows   = (long)in_sizes[0] / 3;   // 8,388,608
    const int ngroups = (int)(rows / 4);         // 4 rows per group
    const int tail    = (int)(rows % 4);

    const int threads = 256;
    int blocks = 2048;                           // ~4 groups/thread at full size
    int maxBlocks = (ngroups + threads - 1) / threads;
    if (maxBlocks < 1) maxBlocks = 1;
    if (blocks > maxBlocks) blocks = maxBlocks;
    if ((size_t)blocks * sizeof(float) > ws_size && ws_size >= sizeof(float))
        blocks = (int)(ws_size / sizeof(float));

    loss_partial_kernel<<<blocks, threads, 0, stream>>>(pred, lab, partials,
                                                        ngroups, tail, rows);
    loss_final_kernel<<<1, 32, 0, stream>>>(partials, blocks, out);
}